// FilterDetections_55336358642130
// MI455X (gfx1250) — compile-verified
//
#include <hip/hip_runtime.h>
#include <stdint.h>

// ---------------- problem constants (match reference) ----------------
#define NPTS        50000
#define NCLASS      80
#define MAX_DET     300
#define SCORE_THR   0.05f
#define NMS_THR     0.5f
#define NEGV        (-1e30f)
#define THREADS     1024   // 32 wave32 waves per workgroup

#define NQ4         (NPTS / 4)             // 12500 float4 score quads
#define MCAND       (NCLASS * MAX_DET)     // 24000 candidates / batch
#define MQ4         (MCAND / 4)            // 6000 float4 quads

// dynamic LDS layouts
#define SMEM_NMS   (NPTS * 4 + THREADS * 4 + THREADS * 4)       // 208192 B (<= 320KB WGP LDS)
#define SMEM_TOPK  (MCAND * 4 + THREADS * 4 + THREADS * 4)      // 104192 B

typedef __attribute__((ext_vector_type(4))) unsigned int u32x4;
typedef __attribute__((ext_vector_type(4))) int          i32x4;
typedef __attribute__((ext_vector_type(8))) int          i32x8;

// ---------------- TDM: 2-D strided tile -> LDS (wave-level, one issue) ----------------
// tile (tile0 x tile1) elements of 4B; row stride = stride0 elements; dest LDS contiguous.
__device__ __forceinline__ void tdm_load_f32_2d(uint64_t ga, uint32_t lds,
                                                uint32_t tile0, uint32_t tile1,
                                                uint32_t td0, uint32_t td1,
                                                uint32_t stride0) {
  u32x4 g0;
  g0[0] = 1u;                                                 // count=1, user descriptor
  g0[1] = lds;                                                // lds_addr (bytes)
  g0[2] = (uint32_t)ga;                                       // global_addr[31:0]
  g0[3] = (uint32_t)((ga >> 32) & 0x1FFFFFFu) | (2u << 30);   // addr[56:32] | type=2

  i32x8 g1;
  g1[0] = (int)(2u << 16);                          // data_size=4B, workgroup_mask=0
  g1[1] = (int)(td0 << 16);                         // tensor_dim0[15:0]
  g1[2] = (int)((td0 >> 16) | (td1 << 16));         // tensor_dim0[31:16] | tensor_dim1[15:0]
  g1[3] = (int)((td1 >> 16) | (tile0 << 16));       // tensor_dim1[31:16] | tile_dim0
  g1[4] = (int)(tile1 & 0xFFFFu);                   // tile_dim1 (0 = unused/1-D), tile_dim2=0
  g1[5] = (int)stride0;                             // tensor_dim0_stride[31:0] (elements)
  g1[6] = 0;                                        // stride hi | tensor_dim1_stride lo
  g1[7] = 0;

  i32x4 z4 = {0, 0, 0, 0};
#if defined(__clang_major__) && (__clang_major__ >= 23)
  i32x8 z8 = {0, 0, 0, 0, 0, 0, 0, 0};
  __builtin_amdgcn_tensor_load_to_lds(g0, g1, z4, z4, z8, 0);
#else
  __builtin_amdgcn_tensor_load_to_lds(g0, g1, z4, z4, 0);
#endif
}

// ---------------- wave32 + LDS block argmax (value, index) ----------------
__device__ __forceinline__ void block_argmax(float v, int i,
                                             float* rv, int* ri, int t,
                                             float& outv, int& outi) {
  #pragma unroll
  for (int m = 16; m > 0; m >>= 1) {              // wave32 butterfly
    float ov = __shfl_xor(v, m, 32);
    int   oi = __shfl_xor(i, m, 32);
    if (ov > v || (ov == v && oi < i)) { v = ov; i = oi; }
  }
  const int wave = t >> 5, lane = t & 31;
  if (lane == 0) { rv[wave] = v; ri[wave] = i; }
  __syncthreads();
  if (wave == 0) {
    const int nw = blockDim.x >> 5;
    float v2 = (lane < nw) ? rv[lane] : NEGV;
    int   i2 = (lane < nw) ? ri[lane] : 0;
    #pragma unroll
    for (int m = 16; m > 0; m >>= 1) {
      float ov = __shfl_xor(v2, m, 32);
      int   oi = __shfl_xor(i2, m, 32);
      if (ov > v2 || (ov == v2 && oi < i2)) { v2 = ov; i2 = oi; }
    }
    if (lane == 0) { rv[0] = v2; ri[0] = i2; }
  }
  __syncthreads();
  outv = rv[0]; outi = ri[0];
}

// in-quad argmax update, ascending index tie-break
__device__ __forceinline__ void quad_max(float4 w, int base, float& bv, int& bi) {
  if (w.x > bv) { bv = w.x; bi = base + 0; }
  if (w.y > bv) { bv = w.y; bi = base + 1; }
  if (w.z > bv) { bv = w.z; bi = base + 2; }
  if (w.w > bv) { bv = w.w; bi = base + 3; }
}

// ---------------- kernel 1: per-(batch,class) greedy NMS ----------------
extern "C" __global__ void nms_per_class(const float* __restrict__ boxes,   // (B,N,4)
                                         const float* __restrict__ cls,    // (B,N,C)
                                         float* __restrict__ cand_score,   // (B,C,MAX_DET)
                                         int*   __restrict__ cand_idx) {   // (B,C,MAX_DET)
  extern __shared__ float smem[];
  float*  s   = smem;                          // NPTS score column (200 KB LDS)
  float4* s4  = (float4*)smem;                 // 128-bit LDS view (ds_*_b128)
  float*  rv  = smem + NPTS;                   // THREADS reduction values
  int*    ri  = (int*)(rv + THREADS);          // THREADS reduction indices

  const int b = blockIdx.x / NCLASS;
  const int c = blockIdx.x % NCLASS;
  const int t = threadIdx.x;

  const float4* __restrict__ bxv = (const float4*)(boxes + (size_t)b * NPTS * 4);

  if (t < 32) {
    // wave 0: TDM strided-column gather classification[b, :, c] -> LDS
    uint64_t ga = (uint64_t)(uintptr_t)cls +
                  ((uint64_t)b * NPTS * NCLASS + (uint64_t)c) * 4ull;
    tdm_load_f32_2d(ga, (uint32_t)(uintptr_t)(void*)s,
                    /*tile*/ 1u, NPTS, /*tensor*/ 1u, NPTS, /*stride0*/ NCLASS);
    __builtin_amdgcn_s_wait_tensorcnt(0);      // column resident in LDS
  } else {
    // other 31 waves: pre-warm box array into L2 (global_prefetch_b8)
    for (int n = (t - 32) * 16; n < NPTS; n += (THREADS - 32) * 16)
      __builtin_prefetch((const void*)(bxv + n), 0, 1);
  }
  __syncthreads();

  // threshold pass (128-bit LDS ops):  s[n] = score>THR ? score : NEG
  for (int q = t; q < NQ4; q += THREADS) {
    float4 w = s4[q];
    w.x = (w.x > SCORE_THR) ? w.x : NEGV;
    w.y = (w.y > SCORE_THR) ? w.y : NEGV;
    w.z = (w.z > SCORE_THR) ? w.z : NEGV;
    w.w = (w.w > SCORE_THR) ? w.w : NEGV;
    s4[q] = w;
  }
  __syncthreads();

  float* cs = cand_score + ((size_t)b * NCLASS + c) * MAX_DET;
  int*   ci = cand_idx   + ((size_t)b * NCLASS + c) * MAX_DET;

  for (int det = 0; det < MAX_DET; ++det) {
    // argmax sweep over LDS column, 4 elements per ds_load_b128
    float bv = NEGV; int bi = 0;
    for (int q = t; q < NQ4; q += THREADS) quad_max(s4[q], q * 4, bv, bi);

    float sv; int si;
    block_argmax(bv, bi, rv, ri, t, sv, si);

    if (sv <= NEGV * 0.5f) {                   // uniform: nothing valid remains
      for (int k = det + t; k < MAX_DET; k += THREADS) { cs[k] = NEGV; ci[k] = 0; }
      break;
    }
    if (t == 0) { cs[det] = sv; ci[det] = si; s[si] = NEGV; }

    const float4 sb = bxv[si];                 // broadcast load of selected box
    const float  sa = (sb.z - sb.x) * (sb.w - sb.y);
    __syncthreads();

    // suppression sweep: IoU(selected, n) > 0.5 -> s[n] = NEG
    for (int q = t; q < NQ4; q += THREADS) {
      float4 w = s4[q];
      bool any = false;
      #pragma unroll
      for (int k = 0; k < 4; ++k) {
        float v = (k == 0) ? w.x : (k == 1) ? w.y : (k == 2) ? w.z : w.w;
        if (v > NEGV * 0.5f) {
          const float4 ob = bxv[q * 4 + k];    // global_load_b128, L2-resident
          float x1 = fmaxf(sb.x, ob.x), y1 = fmaxf(sb.y, ob.y);
          float x2 = fminf(sb.z, ob.z), y2 = fminf(sb.w, ob.w);
          float inter = fmaxf(x2 - x1, 0.f) * fmaxf(y2 - y1, 0.f);
          float a2 = (ob.z - ob.x) * (ob.w - ob.y);
          float uni = sa + a2 - inter;
          float iou = (uni > 0.f) ? inter / uni : 0.f;
          if (iou > NMS_THR) {
            if (k == 0) w.x = NEGV; else if (k == 1) w.y = NEGV;
            else if (k == 2) w.z = NEGV; else w.w = NEGV;
            any = true;
          }
        }
      }
      if (any) s4[q] = w;                      // ds_store_b128 only when changed
    }
    __syncthreads();
  }
}

// ---------------- kernel 2: global top-300 per batch + output formatting ----------------
extern "C" __global__ void topk_out(const float* __restrict__ boxes,
                                    const float* __restrict__ cand_score,
                                    const int*   __restrict__ cand_idx,
                                    float* __restrict__ out_boxes,   // (B,MAX_DET,4)
                                    float* __restrict__ out_scores,  // (B,MAX_DET)
                                    float* __restrict__ out_labels) {// (B,MAX_DET) as f32
  extern __shared__ float smem[];
  float*  ss  = smem;                          // MCAND mutable candidate scores (96 KB)
  float4* ss4 = (float4*)smem;
  float*  rv  = smem + MCAND;
  int*    ri  = (int*)(rv + THREADS);

  const int b = blockIdx.x, t = threadIdx.x;
  const int*    cib = cand_idx + (size_t)b * MCAND;
  const float4* bxv = (const float4*)(boxes + (size_t)b * NPTS * 4);

  if (t < 32) {
    // wave 0: TDM 1-D contiguous load of this batch's candidate scores -> LDS
    uint64_t ga = (uint64_t)(uintptr_t)cand_score + (uint64_t)b * MCAND * 4ull;
    tdm_load_f32_2d(ga, (uint32_t)(uintptr_t)(void*)ss,
                    /*tile*/ MCAND, 0u, /*tensor*/ MCAND, 1u, /*stride0*/ MCAND);
    __builtin_amdgcn_s_wait_tensorcnt(0);
  }
  __syncthreads();

  for (int det = 0; det < MAX_DET; ++det) {
    float bv = NEGV; int bi = 0;
    for (int q = t; q < MQ4; q += THREADS) quad_max(ss4[q], q * 4, bv, bi);

    float sv; int si;
    block_argmax(bv, bi, rv, ri, t, sv, si);

    if (t == 0) {
      ss[si] = NEGV;
      const bool valid = sv > NEGV * 0.5f;
      const int lab  = si / MAX_DET;
      const int bidx = cib[si];
      const float4 bb = bxv[bidx];
      const size_t ob = ((size_t)b * MAX_DET + det) * 4;
      out_boxes[ob + 0] = valid ? bb.x : -1.f;
      out_boxes[ob + 1] = valid ? bb.y : -1.f;
      out_boxes[ob + 2] = valid ? bb.z : -1.f;
      out_boxes[ob + 3] = valid ? bb.w : -1.f;
      out_scores[(size_t)b * MAX_DET + det] = valid ? sv : -1.f;
      out_labels[(size_t)b * MAX_DET + det] = valid ? (float)lab : -1.f;
    }
    __syncthreads();
  }
}

// ---------------- host launch ----------------
extern "C" void kernel_launch(void* const* d_in, const int* in_sizes, int n_in,
                              void* d_out, int out_size, void* d_ws, size_t ws_size,
                              hipStream_t stream) {
  const float* boxes = (const float*)d_in[0];          // (B,N,4)
  const float* cls   = (const float*)d_in[1];          // (B,N,C)
  const int B = in_sizes[0] / (NPTS * 4);

  float* cand_score = (float*)d_ws;                                    // B*C*300 f32
  int*   cand_idx   = (int*)((char*)d_ws +
                      (size_t)B * MCAND * sizeof(float));              // B*C*300 i32

  float* out        = (float*)d_out;
  float* out_boxes  = out;                                   // B*300*4
  float* out_scores = out + (size_t)B * MAX_DET * 4;         // B*300
  float* out_labels = out + (size_t)B * MAX_DET * 5;         // B*300

  (void)hipFuncSetAttribute((const void*)nms_per_class,
                            hipFuncAttributeMaxDynamicSharedMemorySize, SMEM_NMS);
  (void)hipFuncSetAttribute((const void*)topk_out,
                            hipFuncAttributeMaxDynamicSharedMemorySize, SMEM_TOPK);

  nms_per_class<<<B * NCLASS, THREADS, SMEM_NMS, stream>>>(boxes, cls, cand_score, cand_idx);
  topk_out<<<B, THREADS, SMEM_TOPK, stream>>>(boxes, cand_score, cand_idx,
                                              out_boxes, out_scores, out_labels);
  (void)n_in; (void)out_size; (void)ws_size;
}